// Snack_80513456931391
// MI455X (gfx1250) — compile-verified
//
#include <hip/hip_runtime.h>
#include <hip/hip_bf16.h>
#include <math.h>

#define NUM_AA 20
#define FD 16

typedef __attribute__((ext_vector_type(2))) float v2f;
typedef __attribute__((ext_vector_type(8))) float v8f;

// -----------------------------------------------------------------------------
// Kernel 1: build the 20x20 tanh(delta^T M delta) lookup table with f32 WMMA.
//   G = Fp * M * Fp^T   (Fp = features padded 20x16 -> 32x16)
//   dist(i,j) = G[i][i] - G[i][j] - G[j][i] + G[j][j]
// Single wave (32 threads): EXEC is all ones for every WMMA, no inter-wave
// barriers needed (single-wave workgroup barriers lower to S_NOP).
// -----------------------------------------------------------------------------
__global__ __launch_bounds__(32) void snack_table_kernel(
    const float* __restrict__ F, const float* __restrict__ M,
    float* __restrict__ table /* 400 floats */) {
  __shared__ float sF[32][FD];   // padded features
  __shared__ float sM[FD][FD];
  __shared__ float sQ[32][FD];   // Q = Fp * M
  __shared__ float sG[32][32];   // G = Q * Fp^T

  const int lane = threadIdx.x;      // 0..31
  const int half = lane >> 4;        // 0: lanes 0-15, 1: lanes 16-31
  const int lm   = lane & 15;

  // Cooperative load of F (zero-padded rows 20..31) and M into LDS.
  for (int e = lane; e < 32 * FD; e += 32) {
    int r = e >> 4, c = e & 15;
    sF[r][c] = (r < NUM_AA) ? F[r * FD + c] : 0.0f;
  }
  for (int e = lane; e < FD * FD; e += 32) sM[e >> 4][e & 15] = M[e];
  __syncthreads();

  // ---- Stage 1: Q = Fp(32x16) * M(16x16), two 16x16 row tiles -------------
  // A 16x4 layout: lanes 0-15 hold K=4k,4k+1 in v0/v1; lanes 16-31 K=4k+2,4k+3.
  // B 4x16 layout: lane = column; lanes 0-15 rows 4k,4k+1; lanes 16-31 rows +2.
  for (int rb = 0; rb < 2; ++rb) {
    v8f c = {};
    for (int k = 0; k < 4; ++k) {
      const int kk = 4 * k + 2 * half;
      v2f a, b;
      a.x = sF[rb * 16 + lm][kk];
      a.y = sF[rb * 16 + lm][kk + 1];
      b.x = sM[kk][lm];
      b.y = sM[kk + 1][lm];
      c = __builtin_amdgcn_wmma_f32_16x16x4_f32(
          /*neg_a=*/false, a, /*neg_b=*/false, b,
          /*c_mod=*/(short)0, c, /*reuse_a=*/false, /*reuse_b=*/false);
    }
    // C/D layout: VGPR r holds row (r + 8*half), column = lane%16.
    for (int r = 0; r < 8; ++r) sQ[rb * 16 + r + 8 * half][lm] = c[r];
  }
  __syncthreads();

  // ---- Stage 2: G = Q(32x16) * Fp^T(16x32), four 16x16 tiles --------------
  // B = Fp^T chunk: B[kr][n] = Fp[cb*16+n][kr]  (Fp rows become columns).
  for (int rb = 0; rb < 2; ++rb) {
    for (int cb = 0; cb < 2; ++cb) {
      v8f c = {};
      for (int k = 0; k < 4; ++k) {
        const int kk = 4 * k + 2 * half;
        v2f a, b;
        a.x = sQ[rb * 16 + lm][kk];
        a.y = sQ[rb * 16 + lm][kk + 1];
        b.x = sF[cb * 16 + lm][kk];
        b.y = sF[cb * 16 + lm][kk + 1];
        c = __builtin_amdgcn_wmma_f32_16x16x4_f32(
            false, a, false, b, (short)0, c, false, false);
      }
      for (int r = 0; r < 8; ++r)
        sG[rb * 16 + r + 8 * half][cb * 16 + lm] = c[r];
    }
  }
  __syncthreads();

  // ---- Table: tanh of the quadratic form assembled from G -----------------
  for (int p = lane; p < NUM_AA * NUM_AA; p += 32) {
    const int i = p / NUM_AA, j = p % NUM_AA;
    const float d = sG[i][i] - sG[i][j] - sG[j][i] + sG[j][j];
    table[p] = tanhf(d);
  }
}

// -----------------------------------------------------------------------------
// Kernel 2: memory-bound gather. out[p] = table[idx_i[p]*20 + idx_j[p]].
// int4/float4 vectorized (16B vmem ops), table staged in LDS.
// Roofline: ~100 MB of traffic -> ~4.3 us at 23.3 TB/s HBM (L2-resident even).
// -----------------------------------------------------------------------------
__global__ __launch_bounds__(256) void snack_gather_kernel(
    const int* __restrict__ idx_i, const int* __restrict__ idx_j,
    const float* __restrict__ table, float* __restrict__ out,
    int n, int n4) {
  __shared__ float stab[NUM_AA * NUM_AA];
  for (int e = threadIdx.x; e < NUM_AA * NUM_AA; e += 256) stab[e] = table[e];
  __syncthreads();

  const int gid = blockIdx.x * blockDim.x + threadIdx.x;
  if (gid < n4) {
    const int4 a = ((const int4*)idx_i)[gid];
    const int4 b = ((const int4*)idx_j)[gid];
    float4 r;
    r.x = stab[a.x * NUM_AA + b.x];
    r.y = stab[a.y * NUM_AA + b.y];
    r.z = stab[a.z * NUM_AA + b.z];
    r.w = stab[a.w * NUM_AA + b.w];
    ((float4*)out)[gid] = r;
  }
  // Scalar tail (n not divisible by 4) — negligible, one thread.
  if (gid == 0) {
    for (int e = n4 * 4; e < n; ++e)
      out[e] = stab[idx_i[e] * NUM_AA + idx_j[e]];
  }
}

extern "C" void kernel_launch(void* const* d_in, const int* in_sizes, int n_in,
                              void* d_out, int out_size, void* d_ws, size_t ws_size,
                              hipStream_t stream) {
  (void)n_in; (void)out_size; (void)ws_size;
  const float* F   = (const float*)d_in[0];  // [20,16]
  const float* M   = (const float*)d_in[1];  // [16,16]
  const int*   ii  = (const int*)d_in[2];    // [P]
  const int*   jj  = (const int*)d_in[3];    // [P]
  float*       out = (float*)d_out;          // [P]
  float*       tab = (float*)d_ws;           // 400 floats of scratch

  snack_table_kernel<<<1, 32, 0, stream>>>(F, M, tab);

  const int n  = in_sizes[2];
  const int n4 = n >> 2;
  const int blocks = (n4 + 255) / 256 > 0 ? (n4 + 255) / 256 : 1;
  snack_gather_kernel<<<blocks, 256, 0, stream>>>(ii, jj, tab, out, n, n4);
}